// G_process_50629074485344
// MI455X (gfx1250) — compile-verified
//
#include <hip/hip_runtime.h>
#include <hip/hip_bf16.h>

#define NE_N    50000
#define NPROC_N 25000
#define E_N     200000
#define DD      128
#define LL      3

typedef __attribute__((ext_vector_type(2))) float v2f;
typedef __attribute__((ext_vector_type(8))) float v8f;

// ---------------------------------------------------------------------------
// GEMM: H[N x 128] = X[N x 128] @ W[128 x 128], fp32 via V_WMMA_F32_16X16X4_F32
// Block = 256 threads = 8 waves. Each wave: 16 rows x 128 cols.
//
// W staged in LDS in a pair-interleaved XOR-swizzled layout:
//   element (k, n) -> float index  (k>>1)*256 + (n ^ (16*((k>>1)&1)))*2 + (k&1)
// so each WMMA B-operand pair {K, K+1} is one conflict-free ds_load_b64.
// Staging uses gfx1250 async global->LDS b32 copies (arbitrary per-lane LDS
// destination, tracked by ASYNCcnt).
//
// A layout (16x4 f32): lane L holds M = L&15, K = g + 2*(L>>4) in VGPR g.
// B layout (4x16 f32): lane L holds N = L&15, K = g + 2*(L>>4) in VGPR g.
// C/D layout (16x16 f32): lane L, VGPR g -> M = g + 8*(L>>4), N = L&15.
// ---------------------------------------------------------------------------
__global__ __launch_bounds__(256)
void gemm_xw_wmma(const float* __restrict__ X, const float* __restrict__ W,
                  float* __restrict__ H, int N) {
  __shared__ __align__(16) float sB[DD * DD];   // 64 KB swizzled W

  // Async swizzled stage of W into LDS: 256 thr x 64 iters x 4B = 64 KB.
  {
    unsigned lds_base = (unsigned)(unsigned long long)(uintptr_t)&sB[0];
    unsigned long long gbase = (unsigned long long)(uintptr_t)W;
    for (int it = 0; it < 64; ++it) {
      unsigned eidx  = (unsigned)it * 256u + threadIdx.x;   // element index
      unsigned k     = eidx >> 7;
      unsigned n     = eidx & 127u;
      unsigned kpair = k >> 1;
      unsigned f     = n ^ ((kpair & 1u) << 4);
      unsigned didx  = kpair * 256u + f * 2u + (k & 1u);
      unsigned ldsaddr = lds_base + didx * 4u;
      unsigned goff    = eidx * 4u;
      asm volatile("global_load_async_to_lds_b32 %0, %1, %2"
                   :: "v"(ldsaddr), "v"(goff), "s"(gbase)
                   : "memory");
    }
    asm volatile("s_wait_asynccnt 0x0" ::: "memory");
  }
  __syncthreads();

  const int wave = threadIdx.x >> 5;
  const int lane = threadIdx.x & 31;
  const int mrow = lane & 15;
  const int lh   = lane >> 4;
  const int col  = lane & 15;

  const long row0 = ((long)blockIdx.x * 8 + wave) * 16;
  if (row0 >= N) return;   // uniform per-wave: surviving waves keep EXEC = all-1s

  long rA = row0 + mrow; if (rA > N - 1) rA = N - 1;   // clamp tail loads
  const float* __restrict__ xrow = X + rA * DD;

  // Per-lane B base offsets (floats): addr(k2, nt) = k2*256 + boff[nt]
  // where boff[nt] = lh*256 + col*2 + ((nt<<5) ^ (lh<<5)).
  const float* bp[8];
  {
    unsigned laneBase = ((unsigned)lh << 8) + ((unsigned)col << 1);
    #pragma unroll
    for (int nt = 0; nt < 8; ++nt)
      bp[nt] = sB + (laneBase + ((((unsigned)nt << 5)) ^ ((unsigned)lh << 5)));
  }

  v8f c[8];
  #pragma unroll
  for (int nt = 0; nt < 8; ++nt)
    #pragma unroll
    for (int g = 0; g < 8; ++g) c[nt][g] = 0.0f;

  for (int k = 0; k < DD; k += 4) {
    const int k2 = k >> 1;           // even; kpair for lh=0, kpair-1... folded in bp
    v2f a;
    a.x = xrow[k + 2 * lh];
    a.y = xrow[k + 2 * lh + 1];
    #pragma unroll
    for (int nt = 0; nt < 8; ++nt) {
      v2f b = *(const v2f*)(bp[nt] + (unsigned)k2 * 256u);   // one ds_load_b64
      c[nt] = __builtin_amdgcn_wmma_f32_16x16x4_f32(
          false, a, false, b, (short)0, c[nt], false, false);
    }
  }

  #pragma unroll
  for (int nt = 0; nt < 8; ++nt) {
    #pragma unroll
    for (int g = 0; g < 8; ++g) {
      long r = row0 + g + 8 * lh;
      if (r < N) H[r * DD + nt * 16 + col] = c[nt][g];
    }
  }
}

// wsa[i] = sum_j Ws[i,j]*as[j];  wda[i] = sum_j Wd[i,j]*ad[j]   (one block, 128 thr)
__global__ void fused_att_kernel(const float* __restrict__ Ws, const float* __restrict__ as_,
                                 const float* __restrict__ Wd, const float* __restrict__ ad_,
                                 float* __restrict__ wsa, float* __restrict__ wda) {
  int i = threadIdx.x;
  float s1 = 0.f, s2 = 0.f;
  for (int j = 0; j < DD; ++j) {
    s1 += Ws[i * DD + j] * as_[j];
    s2 += Wd[i * DD + j] * ad_[j];
  }
  wsa[i] = s1; wda[i] = s2;
}

// out[n] = dot(X[n,:], v)  — one wave32 per node, float4 per lane + shuffle reduce
__global__ void rowdot_kernel(const float* __restrict__ X, const float* __restrict__ v,
                              float* __restrict__ out, int N) {
  int node = (int)((blockIdx.x * (long)blockDim.x + threadIdx.x) >> 5);
  int lane = threadIdx.x & 31;
  if (node >= N) return;
  float4 a = ((const float4*)(X + (long)node * DD))[lane];
  float4 b = ((const float4*)v)[lane];
  float s = a.x * b.x + a.y * b.y + a.z * b.z + a.w * b.w;
  for (int off = 16; off > 0; off >>= 1) s += __shfl_down(s, off);
  if (lane == 0) out[node] = s;
}

__global__ void fill_zero_kernel(float* __restrict__ p, long n) {
  long i = blockIdx.x * (long)blockDim.x + threadIdx.x;
  if (i < n) p[i] = 0.0f;
}

// x_next[n, c] = b0[c] + b1[c]  (sum of the two biases targeting this node type)
__global__ void init_next_kernel(float* __restrict__ x, const float* __restrict__ b0,
                                 const float* __restrict__ b1, long n_elems) {
  long i = blockIdx.x * (long)blockDim.x + threadIdx.x;
  if (i < n_elems) {
    int c = (int)(i & (DD - 1));
    x[i] = b0[c] + b1[c];
  }
}

// pass 1: per-dst max of relu(es[src]+ed[dst]); logits >= 0 so uint-ordered atomicMax
__global__ void edge_max_kernel(const int* __restrict__ src, const int* __restrict__ dst,
                                const float* __restrict__ es, const float* __restrict__ ed,
                                float* __restrict__ m, int E) {
  int e = blockIdx.x * blockDim.x + threadIdx.x;
  if (e >= E) return;
  int d = dst[e];
  float l = es[src[e]] + ed[d];
  l = l > 0.f ? l : 0.f;
  atomicMax((unsigned int*)(m + d), __float_as_uint(l));
}

// pass 2: denom[dst] += exp(logit - m[dst])
__global__ void edge_denom_kernel(const int* __restrict__ src, const int* __restrict__ dst,
                                  const float* __restrict__ es, const float* __restrict__ ed,
                                  const float* __restrict__ m, float* __restrict__ denom, int E) {
  int e = blockIdx.x * blockDim.x + threadIdx.x;
  if (e >= E) return;
  int d = dst[e];
  float l = es[src[e]] + ed[d];
  l = l > 0.f ? l : 0.f;
  unsafeAtomicAdd(denom + d, expf(l - m[d]));
}

// pass 3: out[dst,:] += alpha * hs[src,:]  — one wave32 per edge, 4 floats/lane
__global__ void edge_scatter_kernel(const int* __restrict__ src, const int* __restrict__ dst,
                                    const float* __restrict__ es, const float* __restrict__ ed,
                                    const float* __restrict__ m, const float* __restrict__ denom,
                                    const float* __restrict__ hs, float* __restrict__ out, int E) {
  int e = (int)((blockIdx.x * (long)blockDim.x + threadIdx.x) >> 5);
  int lane = threadIdx.x & 31;
  if (e >= E) return;
  int s = src[e], d = dst[e];
  float l = es[s] + ed[d];
  l = l > 0.f ? l : 0.f;
  float alpha = expf(l - m[d]) / fmaxf(denom[d], 1e-16f);
  float4 h = ((const float4*)(hs + (long)s * DD))[lane];
  float* o = out + (long)d * DD + lane * 4;
  unsafeAtomicAdd(o + 0, alpha * h.x);
  unsafeAtomicAdd(o + 1, alpha * h.y);
  unsafeAtomicAdd(o + 2, alpha * h.z);
  unsafeAtomicAdd(o + 3, alpha * h.w);
}

// column sums: block of 128 threads covers 64 rows; one atomicAdd per (block, col)
__global__ __launch_bounds__(128)
void colsum_kernel(const float* __restrict__ X, float* __restrict__ colsum, int N) {
  int c = threadIdx.x;
  long r0 = (long)blockIdx.x * 64;
  float s = 0.f;
  for (int i = 0; i < 64; ++i) {
    long r = r0 + i;
    if (r < N) s += X[r * DD + c];
  }
  unsafeAtomicAdd(colsum + c, s);
}

__global__ __launch_bounds__(128)
void final_kernel(const float* __restrict__ colsum, const float* __restrict__ linW,
                  const float* __restrict__ linb, float* __restrict__ out) {
  __shared__ float p0[DD], p1[DD];
  int i = threadIdx.x;
  p0[i] = colsum[i] * linW[i * 2 + 0];
  p1[i] = colsum[i] * linW[i * 2 + 1];
  __syncthreads();
  for (int off = 64; off > 0; off >>= 1) {
    if (i < off) { p0[i] += p0[i + off]; p1[i] += p1[i + off]; }
    __syncthreads();
  }
  if (i == 0) {
    const float inv = 1.0f / (float)(NE_N + NPROC_N);
    out[0] = p0[0] * inv + linb[0];
    out[1] = p1[0] * inv + linb[1];
  }
}

// ---------------------------------------------------------------------------
extern "C" void kernel_launch(void* const* d_in, const int* in_sizes, int n_in,
                              void* d_out, int out_size, void* d_ws, size_t ws_size,
                              hipStream_t stream) {
  const float* x_elem  = (const float*)d_in[0];
  const float* x_proc  = (const float*)d_in[1];
  const float* W_src   = (const float*)d_in[2];
  const float* W_dst   = (const float*)d_in[3];
  const float* att_src = (const float*)d_in[4];
  const float* att_dst = (const float*)d_in[5];
  const float* bias    = (const float*)d_in[6];
  const float* lin_W   = (const float*)d_in[7];
  const float* lin_b   = (const float*)d_in[8];
  const int* idx[8];
  for (int i = 0; i < 8; ++i) idx[i] = (const int*)d_in[9 + i];

  // ---- carve workspace (floats, 256-float aligned regions) ----
  float* base = (float*)d_ws;
  size_t off = 0;
  auto carve = [&](size_t nfloats) -> float* {
    float* p = base + off;
    off += (nfloats + 255) & ~(size_t)255;
    return p;
  };
  float* xeA   = carve((size_t)NE_N * DD);
  float* xeB   = carve((size_t)NE_N * DD);
  float* xpA   = carve((size_t)NPROC_N * DD);
  float* xpB   = carve((size_t)NPROC_N * DD);
  float* hs    = carve((size_t)NE_N * DD);   // reused per edge type
  float* es    = carve(NE_N);
  float* ed    = carve(NE_N);
  float* mbuf  = carve(NE_N);
  float* dbuf  = carve(NE_N);
  float* wsa   = carve(DD);
  float* wda   = carve(DD);
  float* csum  = carve(DD);
  (void)ws_size; (void)in_sizes; (void)n_in; (void)out_size;

  hipMemcpyAsync(xeA, x_elem, (size_t)NE_N * DD * sizeof(float),
                 hipMemcpyDeviceToDevice, stream);
  hipMemcpyAsync(xpA, x_proc, (size_t)NPROC_N * DD * sizeof(float),
                 hipMemcpyDeviceToDevice, stream);

  float* xe_cur = xeA; float* xe_next = xeB;
  float* xp_cur = xpA; float* xp_next = xpB;

  for (int l = 0; l < LL; ++l) {
    // init next buffers with summed biases per dst type: xe <- ee+pe, xp <- ep+pp
    {
      long ne = (long)NE_N * DD;
      init_next_kernel<<<(unsigned)((ne + 255) / 256), 256, 0, stream>>>(
          xe_next, bias + ((size_t)l * 4 + 0) * DD, bias + ((size_t)l * 4 + 3) * DD, ne);
      long np = (long)NPROC_N * DD;
      init_next_kernel<<<(unsigned)((np + 255) / 256), 256, 0, stream>>>(
          xp_next, bias + ((size_t)l * 4 + 1) * DD, bias + ((size_t)l * 4 + 2) * DD, np);
    }

    for (int t = 0; t < 4; ++t) {
      const float* xs; int Ns;
      const float* xd; int Nd; float* xnext_d;
      if (t < 2) { xs = xe_cur; Ns = NE_N; } else { xs = xp_cur; Ns = NPROC_N; }
      switch (t) {
        case 0: xd = xe_cur; Nd = NE_N;    xnext_d = xe_next; break;   // e->e
        case 1: xd = xp_cur; Nd = NPROC_N; xnext_d = xp_next; break;   // e->p
        case 2: xd = xp_cur; Nd = NPROC_N; xnext_d = xp_next; break;   // p->p
        default:xd = xe_cur; Nd = NE_N;    xnext_d = xe_next; break;   // p->e
      }
      const int* s_idx = idx[2 * t];
      const int* d_idx = idx[2 * t + 1];
      size_t wo = (size_t)l * 4 + t;

      fused_att_kernel<<<1, DD, 0, stream>>>(
          W_src + wo * DD * DD, att_src + wo * DD,
          W_dst + wo * DD * DD, att_dst + wo * DD, wsa, wda);

      gemm_xw_wmma<<<(unsigned)((Ns + 127) / 128), 256, 0, stream>>>(
          xs, W_src + wo * DD * DD, hs, Ns);

      rowdot_kernel<<<(unsigned)(((long)Ns * 32 + 255) / 256), 256, 0, stream>>>(xs, wsa, es, Ns);
      rowdot_kernel<<<(unsigned)(((long)Nd * 32 + 255) / 256), 256, 0, stream>>>(xd, wda, ed, Nd);

      fill_zero_kernel<<<(unsigned)((Nd + 255) / 256), 256, 0, stream>>>(mbuf, Nd);
      fill_zero_kernel<<<(unsigned)((Nd + 255) / 256), 256, 0, stream>>>(dbuf, Nd);

      edge_max_kernel<<<(unsigned)((E_N + 255) / 256), 256, 0, stream>>>(
          s_idx, d_idx, es, ed, mbuf, E_N);
      edge_denom_kernel<<<(unsigned)((E_N + 255) / 256), 256, 0, stream>>>(
          s_idx, d_idx, es, ed, mbuf, dbuf, E_N);
      edge_scatter_kernel<<<(unsigned)(((long)E_N * 32 + 255) / 256), 256, 0, stream>>>(
          s_idx, d_idx, es, ed, mbuf, dbuf, hs, xnext_d, E_N);
    }

    float* tmp;
    tmp = xe_cur; xe_cur = xe_next; xe_next = tmp;
    tmp = xp_cur; xp_cur = xp_next; xp_next = tmp;
  }

  fill_zero_kernel<<<1, 256, 0, stream>>>(csum, DD);
  colsum_kernel<<<(unsigned)((NE_N + 63) / 64), DD, 0, stream>>>(xe_cur, csum, NE_N);
  colsum_kernel<<<(unsigned)((NPROC_N + 63) / 64), DD, 0, stream>>>(xp_cur, csum, NPROC_N);
  final_kernel<<<1, DD, 0, stream>>>(csum, lin_W, lin_b, (float*)d_out);
}